// TinyLoRALinear_43293270343961
// MI455X (gfx1250) — compile-verified
//
#include <hip/hip_runtime.h>

typedef float v2f __attribute__((ext_vector_type(2)));
typedef float v8f __attribute__((ext_vector_type(8)));

static constexpr int KDIM = 4096;   // D_IN
static constexpr int NDIM = 4096;   // D_OUT
static constexpr int MDIM = 8192;   // B*T
static constexpr int NSTAGE = KDIM / 32;   // 128 k-tiles
static constexpr int LDSW = 36;     // row stride in floats: 144 B (16B aligned)

// ---------------------------------------------------------------------------
// Kernel 1: M2[o][r] = sum_j U[o][j] * (S @ (sum_u v[u] * P[u]))[j][r]
// ---------------------------------------------------------------------------
__global__ void prep_M_kernel(const float* __restrict__ U, const float* __restrict__ S,
                              const float* __restrict__ P, const float* __restrict__ v,
                              float* __restrict__ M2) {
  int o = blockIdx.x * blockDim.x + threadIdx.x;
  if (o >= NDIM) return;
  float R00 = 0.f, R01 = 0.f, R10 = 0.f, R11 = 0.f;
  for (int u = 0; u < 16; ++u) {
    float vu = v[u];
    R00 += vu * P[u * 4 + 0];
    R01 += vu * P[u * 4 + 1];
    R10 += vu * P[u * 4 + 2];
    R11 += vu * P[u * 4 + 3];
  }
  float SR00 = S[0] * R00 + S[1] * R10;
  float SR01 = S[0] * R01 + S[1] * R11;
  float SR10 = S[2] * R00 + S[3] * R10;
  float SR11 = S[2] * R01 + S[3] * R11;
  float u0 = U[o * 2 + 0], u1 = U[o * 2 + 1];
  M2[o * 2 + 0] = u0 * SR00 + u1 * SR10;
  M2[o * 2 + 1] = u0 * SR01 + u1 * SR11;
}

// ---------------------------------------------------------------------------
// Kernel 2: xV[m][r] = sum_k x[m][k] * Vh[r][k]   (one wave32 per row)
// ---------------------------------------------------------------------------
__global__ void xv_kernel(const float* __restrict__ x, const float* __restrict__ Vh,
                          float* __restrict__ xV) {
  int wave = threadIdx.x >> 5;
  int lane = threadIdx.x & 31;
  int m = blockIdx.x * 8 + wave;
  const float* xr = x + (size_t)m * KDIM;
  float p0 = 0.f, p1 = 0.f;
  for (int k = lane; k < KDIM; k += 32) {
    float xv = xr[k];
    p0 += xv * Vh[k];
    p1 += xv * Vh[KDIM + k];
  }
  for (int s = 16; s > 0; s >>= 1) {
    p0 += __shfl_xor(p0, s);
    p1 += __shfl_xor(p1, s);
  }
  if (lane == 0) {
    xV[m * 2 + 0] = p0;
    xV[m * 2 + 1] = p1;
  }
}

// ---------------------------------------------------------------------------
// Kernel 3: out = x @ W.T + bias + rank-2 delta, fp32 WMMA 16x16x4.
// 128x128 block tile, BK=32, 8 waves; each wave: 64x32 = 4x2 fragments.
// Tile feed: double-buffered global_load_async_to_lds_b128 (ASYNCcnt).
// ---------------------------------------------------------------------------
__global__ __launch_bounds__(256) void gemm_kernel(
    const float* __restrict__ X, const float* __restrict__ W,
    const float* __restrict__ bias, const float* __restrict__ M2,
    const float* __restrict__ xV, float* __restrict__ out) {
  __shared__ float As[2][128][LDSW];
  __shared__ float Bs[2][128][LDSW];

  const int tid  = threadIdx.x;
  const int lane = tid & 31;
  const int wave = tid >> 5;
  const int wm   = wave & 1;    // 2 slabs of 64 rows
  const int wn   = wave >> 1;   // 4 slabs of 32 cols
  const int half = lane >> 4;
  const int lr   = lane & 15;
  const int blockM = blockIdx.y * 128;
  const int blockN = blockIdx.x * 128;

  // --- async-copy addressing: each thread owns 4 x b128 per tile per stage
  const int row0 = tid >> 3;    // 0..31 (rows row0, row0+32, +64, +96)
  const int kg   = tid & 7;     // float4 group within the 32-float k-tile
  const float* Xblk = X + (size_t)blockM * KDIM;
  const float* Wblk = W + (size_t)blockN * KDIM;

  unsigned voff[4];             // global byte offset (per row), stage adds k*4
  unsigned ldsA[2][4], ldsB[2][4];
#pragma unroll
  for (int j = 0; j < 4; ++j) {
    int row = row0 + j * 32;
    voff[j] = (unsigned)(row * KDIM + kg * 4) * 4u;
#pragma unroll
    for (int buf = 0; buf < 2; ++buf) {
      ldsA[buf][j] = (unsigned)(uintptr_t)&As[buf][row][kg * 4];
      ldsB[buf][j] = (unsigned)(uintptr_t)&Bs[buf][row][kg * 4];
    }
  }

#define ISSUE_STAGE(stage)                                                    \
  do {                                                                        \
    const int _buf = (stage) & 1;                                             \
    const unsigned _co = (unsigned)(stage) * 128u; /* 32 floats = 128 B */    \
    _Pragma("unroll")                                                         \
    for (int _j = 0; _j < 4; ++_j) {                                          \
      asm volatile("global_load_async_to_lds_b128 %0, %1, %2"                 \
                   :: "v"(ldsA[_buf][_j]), "v"(voff[_j] + _co), "s"(Xblk)     \
                   : "memory");                                               \
      asm volatile("global_load_async_to_lds_b128 %0, %1, %2"                 \
                   :: "v"(ldsB[_buf][_j]), "v"(voff[_j] + _co), "s"(Wblk)     \
                   : "memory");                                               \
    }                                                                         \
  } while (0)

  v8f acc[4][2];
#pragma unroll
  for (int i = 0; i < 4; ++i)
#pragma unroll
    for (int j = 0; j < 2; ++j)
#pragma unroll
      for (int e = 0; e < 8; ++e) acc[i][j][e] = 0.f;

  ISSUE_STAGE(0);
  ISSUE_STAGE(1);

  for (int i = 0; i < NSTAGE; ++i) {
    // stage i landed when <=8 of our async loads remain (stage i+1 in flight)
    if (i < NSTAGE - 1)
      asm volatile("s_wait_asynccnt 0x8" ::: "memory");
    else
      asm volatile("s_wait_asynccnt 0x0" ::: "memory");
    __syncthreads();

    const int buf = i & 1;
#pragma unroll
    for (int kk = 0; kk < 8; ++kk) {
      const int kb = kk * 4 + half * 2;   // per-lane K pair (ISA 7.12.2 layout)
      v2f a[4], b[2];
#pragma unroll
      for (int fm = 0; fm < 4; ++fm)
        a[fm] = *(const v2f*)&As[buf][wm * 64 + fm * 16 + lr][kb];
#pragma unroll
      for (int fn = 0; fn < 2; ++fn)
        b[fn] = *(const v2f*)&Bs[buf][wn * 32 + fn * 16 + lr][kb];
#pragma unroll
      for (int fm = 0; fm < 4; ++fm)
#pragma unroll
        for (int fn = 0; fn < 2; ++fn)
          acc[fm][fn] = __builtin_amdgcn_wmma_f32_16x16x4_f32(
              false, a[fm], false, b[fn], (short)0, acc[fm][fn], false, false);
    }
    __syncthreads();   // everyone done reading buf before it is overwritten

    if (i + 2 < NSTAGE) ISSUE_STAGE(i + 2);
  }
#undef ISSUE_STAGE

  // Epilogue: bias + rank-2 delta, then store.
#pragma unroll
  for (int fn = 0; fn < 2; ++fn) {
    int n = blockN + wn * 32 + fn * 16 + lr;
    float bn = bias[n];
    float m0 = M2[n * 2 + 0];
    float m1 = M2[n * 2 + 1];
#pragma unroll
    for (int fm = 0; fm < 4; ++fm) {
      int mbase = blockM + wm * 64 + fm * 16 + half * 8;
#pragma unroll
      for (int r = 0; r < 8; ++r) {
        int m = mbase + r;
        float cval = acc[fm][fn][r];
        out[(size_t)m * NDIM + n] =
            cval + bn + xV[m * 2 + 0] * m0 + xV[m * 2 + 1] * m1;
      }
    }
  }
}

// ---------------------------------------------------------------------------
extern "C" void kernel_launch(void* const* d_in, const int* in_sizes, int n_in,
                              void* d_out, int out_size, void* d_ws, size_t ws_size,
                              hipStream_t stream) {
  const float* x  = (const float*)d_in[0];
  const float* W  = (const float*)d_in[1];
  const float* b  = (const float*)d_in[2];
  const float* U  = (const float*)d_in[3];
  const float* S  = (const float*)d_in[4];
  const float* Vh = (const float*)d_in[5];
  const float* P  = (const float*)d_in[6];
  const float* v  = (const float*)d_in[7];
  float* out = (float*)d_out;

  float* M2 = (float*)d_ws;          // [4096][2]
  float* xV = M2 + NDIM * 2;         // [8192][2]  (96 KB total workspace)

  prep_M_kernel<<<NDIM / 256, 256, 0, stream>>>(U, S, P, v, M2);
  xv_kernel<<<MDIM / 8, 256, 0, stream>>>(x, Vh, xV);

  dim3 grid(NDIM / 128, MDIM / 128);  // (32, 64)
  gemm_kernel<<<grid, 256, 0, stream>>>(x, W, b, M2, xV, out);
}